// PhaseEncoding_46651934769191
// MI455X (gfx1250) — compile-verified
//
#include <hip/hip_runtime.h>

typedef __attribute__((ext_vector_type(2))) float v2f;
typedef __attribute__((ext_vector_type(8))) float v8f;

// out[M, D] = x[M, D] + phase[M, P] @ emb[P, D]
// M = S*B = 32768, P = 9, D = 1024.
// Block = 16 rows x 1024 cols, 8 wave32; each wave owns 16x128 (8 WMMA tiles).
// Contraction via V_WMMA_F32_16X16X4_F32, K=9 padded to 12 (3 chunks of 4).
// All global accesses use uniform-base + 32-bit voffset + immediate offsets;
// no divergent branches (half-wave K selection done arithmetically).
__global__ __launch_bounds__(256) void PhaseEncoding_wmma_kernel(
    const float* __restrict__ x,
    const float* __restrict__ ph,
    const float* __restrict__ emb,
    float* __restrict__ out) {
  constexpr int P = 9;
  constexpr int D = 1024;

  const int lane = threadIdx.x & 31;
  const int wave = threadIdx.x >> 5;
  const int m    = lane & 15;        // M index within tile / N index for B,C
  const bool hi  = lane >= 16;       // upper half-wave
  const int hi8  = hi ? 8 : 0;       // C/D row offset for upper half
  const int k0   = hi ? 2 : 0;       // A/B K offset for upper half

  const int row_base = blockIdx.x * 16;
  const int col_base = wave * 128;

  // ---- A fragments: phase rows (16x4 per chunk) ---------------------------
  // lanes 0-15: VGPR0=K(c+0), VGPR1=K(c+1); lanes 16-31: K(c+2), K(c+3)
  const float* phr = ph + (uint32_t)(row_base + m) * P;
  v2f a0, a1, a2;
  a0[0] = phr[k0 + 0]; a0[1] = phr[k0 + 1];
  a1[0] = phr[k0 + 4]; a1[1] = phr[k0 + 5];
  const float a8 = phr[8];                 // valid address for all lanes
  a2[0] = hi ? 0.0f : a8;                  // K=8 valid, K=9..11 padded to 0
  a2[1] = 0.0f;

  // ---- B fragments: emb rows, hoisted for all 8 N-tiles -------------------
  const float* ecb = emb + (uint32_t)(col_base + m) + (uint32_t)k0 * D;
  const float* e8  = emb + (uint32_t)(8 * D + col_base + m);
  v2f b0[8], b1[8], b2[8];
#pragma unroll
  for (int t = 0; t < 8; ++t) {
    b0[t][0] = ecb[t * 16 + 0 * D]; b0[t][1] = ecb[t * 16 + 1 * D];
    b1[t][0] = ecb[t * 16 + 4 * D]; b1[t][1] = ecb[t * 16 + 5 * D];
    const float v8e = e8[t * 16];
    b2[t][0] = hi ? 0.0f : v8e;
    b2[t][1] = 0.0f;
  }

  // ---- streaming loop: C = x tile, D = A*B + C, store ---------------------
  // C/D layout: VGPR r <-> row (row_base + hi8 + r), col = col_base + t*16 + m
  const uint32_t base = (uint32_t)(row_base + hi8) * D + (uint32_t)(col_base + m);
  const float* xb = x + base;
  float* ob       = out + base;

#pragma unroll
  for (int t = 0; t < 8; ++t) {
    v8f c;
#pragma unroll
    for (int r = 0; r < 8; ++r)
      c[r] = __builtin_nontemporal_load(xb + (t * 16 + r * D));

    c = __builtin_amdgcn_wmma_f32_16x16x4_f32(false, a0, false, b0[t], (short)0,
                                              c, false, false);
    c = __builtin_amdgcn_wmma_f32_16x16x4_f32(false, a1, false, b1[t], (short)0,
                                              c, false, false);
    c = __builtin_amdgcn_wmma_f32_16x16x4_f32(false, a2, false, b2[t], (short)0,
                                              c, false, false);

#pragma unroll
    for (int r = 0; r < 8; ++r)
      __builtin_nontemporal_store(c[r], ob + (t * 16 + r * D));
  }
}

extern "C" void kernel_launch(void* const* d_in, const int* in_sizes, int n_in,
                              void* d_out, int out_size, void* d_ws, size_t ws_size,
                              hipStream_t stream) {
  const float* x   = (const float*)d_in[0];  // (S, B, D) f32
  const float* ph  = (const float*)d_in[1];  // (S, B, P) f32
  const float* emb = (const float*)d_in[2];  // (P, D)    f32
  float* out       = (float*)d_out;          // (S, B, D) f32

  const int D = 1024;
  const int M = in_sizes[0] / D;             // S*B = 32768 rows

  dim3 grid(M / 16);
  dim3 block(256);
  PhaseEncoding_wmma_kernel<<<grid, block, 0, stream>>>(x, ph, emb, out);
}